// CRFLayer_49675591746131
// MI455X (gfx1250) — compile-verified
//
#include <hip/hip_runtime.h>

// Problem constants from the reference
constexpr int Bn = 128;
constexpr int Tn = 256;
constexpr int Dn = 1024;
constexpr int Ln = 48;
#define SMALLV (-1000.0f)

typedef __attribute__((ext_vector_type(2))) float v2f;
typedef __attribute__((ext_vector_type(8))) float v8f;

// ---------------------------------------------------------------------------
// Kernel 1: pred[m, n] = sum_k x[m,k] * W[n,k] + b[n], cols >= L-2 = SMALL.
// One wave per 16x48 output slab: all three 16x16 N-tiles fused so each row
// of x is streamed from HBM exactly ONCE (x = 128 MB dominates the roofline;
// W = 192 KB stays L2-resident). Per K-step (K+=4): one A-fragment load
// (contiguous 64b per lane) + three B-fragment loads + three f32 WMMAs.
//
// A fragment (16x4 f32, 2 VGPRs/lane): lanes 0-15 -> M=lane, K=k0+{0,1};
// lanes 16-31 -> M=lane-16, K=k0+{2,3}.
// B fragment (4x16 f32): lanes 0-15 -> N=lane, K=k0+{0,1};
// lanes 16-31 -> N=lane-16, K=k0+{2,3}; B[k,n] = W[n,k].
// C (16x16 f32, 8 VGPRs): VGPR r, lanes 0-15 -> (M=r, N=lane),
//                         lanes 16-31 -> (M=r+8, N=lane-16).
// ---------------------------------------------------------------------------
__global__ __launch_bounds__(32) void crf_gemm_pred_kernel(
    const float* __restrict__ X,     // (B*T, D) row-major
    const float* __restrict__ W,     // (L, D)   row-major
    const float* __restrict__ bias,  // (L)
    float* __restrict__ pred)        // (B*T, L) row-major
{
    const int mtile = blockIdx.x;   // 0 .. (B*T/16 - 1)
    const int lane  = threadIdx.x;  // 0 .. 31
    const int half  = lane >> 4;    // 0 or 1
    const int l16   = lane & 15;

    const int m0 = mtile * 16;

    const float* Arow  = X + (size_t)(m0 + l16) * Dn + 2 * half;
    const float* Brow0 = W + (size_t)( 0 + l16) * Dn + 2 * half;
    const float* Brow1 = W + (size_t)(16 + l16) * Dn + 2 * half;
    const float* Brow2 = W + (size_t)(32 + l16) * Dn + 2 * half;

    v8f c0 = {}, c1 = {}, c2 = {};
#pragma unroll 4
    for (int k0 = 0; k0 < Dn; k0 += 4) {
        v2f a  = *(const v2f*)(Arow  + k0);
        v2f b0 = *(const v2f*)(Brow0 + k0);
        v2f b1 = *(const v2f*)(Brow1 + k0);
        v2f b2 = *(const v2f*)(Brow2 + k0);
        // 8 args: (neg_a, A, neg_b, B, c_mod, C, reuse_a, reuse_b)
        c0 = __builtin_amdgcn_wmma_f32_16x16x4_f32(
                 false, a, false, b0, (short)0, c0, false, false);
        c1 = __builtin_amdgcn_wmma_f32_16x16x4_f32(
                 false, a, false, b1, (short)0, c1, false, false);
        c2 = __builtin_amdgcn_wmma_f32_16x16x4_f32(
                 false, a, false, b2, (short)0, c2, false, false);
    }

    // Epilogue: add bias, mask the last two label columns, store 16x48 slab.
    float* prow = pred + (size_t)m0 * Ln;
#pragma unroll
    for (int nt = 0; nt < 3; ++nt) {
        const v8f&  c   = (nt == 0) ? c0 : (nt == 1) ? c1 : c2;
        const int   col = nt * 16 + l16;
        const float bv  = bias[col];
        const bool  msk = (col >= Ln - 2);
#pragma unroll
        for (int r = 0; r < 8; ++r) {
            const int m = r + 8 * half;
            float v = c[r] + bv;
            if (msk) v = SMALLV;
            prow[(size_t)m * Ln + col] = v;
        }
    }
}

// ---------------------------------------------------------------------------
// Kernel 2: per-batch CRF forward scan + score, one block (2 waves) per batch.
// Thread j (< 48) owns label j; trans column trans[:, j] lives in VGPRs.
// prev[] alpha vector lives in LDS; each step:
//   new[j] = emit[j] + m + log(sum_i exp(prev[i] + T[i,j] - m))
// Runs exactly seq_len+1 steps (rows 1..seq_len+1 of the padded emissions),
// final step uses the synthetic "end" row (SMALL except col L-1 = 0).
// Then the block computes the gold-path score (emission gather + transition
// chain) and atomically accumulates (log_z - real) into out[0].
// ---------------------------------------------------------------------------
__global__ __launch_bounds__(64) void crf_scan_kernel(
    const float* __restrict__ pred,    // (B, T, L)
    const float* __restrict__ trans,   // (L, L)
    const int*   __restrict__ tags,    // (B, T)
    const int*   __restrict__ seqlen,  // (B)
    float* __restrict__ out)           // scalar
{
    const int b = blockIdx.x;
    const int j = threadIdx.x;

    __shared__ float prev[Ln];
    __shared__ float red[64];

    const int sl = seqlen[b];
    const float* pb = pred + (size_t)b * Tn * Ln;

    float tcol[Ln];
    if (j < Ln) {
#pragma unroll
        for (int i = 0; i < Ln; ++i) tcol[i] = trans[i * Ln + j];
        prev[j] = (j == Ln - 2) ? 0.0f : SMALLV;
    }
    __syncthreads();

    for (int t = 1; t <= sl + 1; ++t) {
        float newv = 0.0f;
        if (j < Ln) {
            const float emit = (t <= sl) ? pb[(size_t)(t - 1) * Ln + j]
                                         : ((j == Ln - 1) ? 0.0f : SMALLV);
            float pv[Ln];
#pragma unroll
            for (int i = 0; i < Ln; ++i) pv[i] = prev[i] + tcol[i];
            float m = pv[0];
#pragma unroll
            for (int i = 1; i < Ln; ++i) m = fmaxf(m, pv[i]);
            float s = 0.0f;
#pragma unroll
            for (int i = 0; i < Ln; ++i) s += __expf(pv[i] - m);
            newv = emit + m + __logf(s);
        }
        __syncthreads();
        if (j < Ln) prev[j] = newv;
        __syncthreads();
    }

    // ---- real (gold-path) score, parallel over threads ----
    const int* tg = tags + (size_t)b * Tn;
    float part = 0.0f;
    for (int t = j; t < sl; t += 64)               // emission gather
        part += pb[(size_t)t * Ln + tg[t]];
    for (int k = j; k <= sl; k += 64) {            // transition chain
        const int pk  = (k == 0)  ? (Ln - 2) : tg[k - 1];
        const int pk1 = (k == sl) ? (Ln - 1) : tg[k];
        part += trans[pk * Ln + pk1];
    }
    red[j] = part;
    __syncthreads();

    if (j == 0) {
        float real = 0.0f;
        for (int i = 0; i < 64; ++i) real += red[i];
        float m = prev[0];
        for (int i = 1; i < Ln; ++i) m = fmaxf(m, prev[i]);
        float s = 0.0f;
        for (int i = 0; i < Ln; ++i) s += __expf(prev[i] - m);
        const float logz = m + __logf(s);
        atomicAdd(out, logz - real);
    }
}

__global__ void crf_zero_out_kernel(float* out) {
    if (threadIdx.x == 0 && blockIdx.x == 0) out[0] = 0.0f;
}

extern "C" void kernel_launch(void* const* d_in, const int* in_sizes, int n_in,
                              void* d_out, int out_size, void* d_ws, size_t ws_size,
                              hipStream_t stream) {
    const float* x      = (const float*)d_in[0];  // (B, T, D)
    const float* W      = (const float*)d_in[1];  // (L, D)
    const float* bias   = (const float*)d_in[2];  // (L)
    const float* trans  = (const float*)d_in[3];  // (L, L)
    const int*   tags   = (const int*)d_in[4];    // (B, T)
    const int*   seqlen = (const int*)d_in[5];    // (B)
    float*       out    = (float*)d_out;

    float* pred = (float*)d_ws;   // B*T*L floats = ~6.3 MB scratch

    crf_zero_out_kernel<<<1, 32, 0, stream>>>(out);

    dim3 ggrid((Bn * Tn) / 16);   // one wave per 16-row slab, all 48 cols
    crf_gemm_pred_kernel<<<ggrid, 32, 0, stream>>>(x, W, bias, pred);

    crf_scan_kernel<<<Bn, 64, 0, stream>>>(pred, trans, tags, seqlen, out);
}